// DecoderTransformer_38998303048534
// MI455X (gfx1250) — compile-verified
//
#include <hip/hip_runtime.h>
#include <math.h>

typedef __attribute__((ext_vector_type(2))) float v2f;
typedef __attribute__((ext_vector_type(8))) float v8f;

#define WMMA_F32(a, b, c) \
  __builtin_amdgcn_wmma_f32_16x16x4_f32(false, (a), false, (b), (short)0, (c), false, false)

// ---------------- constants ----------------
#define Vv   32000
#define Tt   1024
#define Dd   1024
#define NHh  16
#define HSs  64
#define Ll   4
#define FFD  4096
#define Bb   4
#define BT   4096   // Bb*Tt
#define EPS  1e-5f

__device__ __forceinline__ float gelu_exact(float x) {
  return 0.5f * x * (1.0f + erff(x * 0.70710678118654752f));
}

// ---------------- embedding: h[bt,d] = emb[x[bt],d] + pos[t,d] ----------------
__global__ __launch_bounds__(256) void embed_kernel(const float* __restrict__ emb,
                                                    const float* __restrict__ pos,
                                                    const int* __restrict__ x,
                                                    float* __restrict__ h) {
  int bt = blockIdx.x;
  int t = bt & (Tt - 1);
  int tok = x[bt];
  const float* e = emb + (size_t)tok * Dd;
  const float* p = pos + (size_t)t * Dd;
  float* o = h + (size_t)bt * Dd;
#pragma unroll
  for (int j = 0; j < 4; ++j) {
    int d = threadIdx.x + j * 256;
    o[d] = e[d] + p[d];
  }
}

// ---------------- layernorm over last dim (D=1024), one row per block ----------------
__global__ __launch_bounds__(256) void ln_kernel(const float* __restrict__ X,
                                                 const float* __restrict__ g,
                                                 const float* __restrict__ bta,
                                                 float* __restrict__ Y) {
  int row = blockIdx.x;
  int tid = threadIdx.x;
  const float* xr = X + (size_t)row * Dd;
  float vals[4];
  float s = 0.f, sq = 0.f;
#pragma unroll
  for (int j = 0; j < 4; ++j) {
    float v = xr[tid + j * 256];
    vals[j] = v;
    s += v;
    sq += v * v;
  }
  __shared__ float rs[256];
  __shared__ float rq[256];
  rs[tid] = s; rq[tid] = sq;
  __syncthreads();
  for (int off = 128; off > 0; off >>= 1) {
    if (tid < off) { rs[tid] += rs[tid + off]; rq[tid] += rq[tid + off]; }
    __syncthreads();
  }
  float mu  = rs[0] * (1.0f / Dd);
  float var = rq[0] * (1.0f / Dd) - mu * mu;
  float inv = rsqrtf(var + EPS);
  float* yr = Y + (size_t)row * Dd;
#pragma unroll
  for (int j = 0; j < 4; ++j) {
    int d = tid + j * 256;
    yr[d] = (vals[j] - mu) * inv * g[d] + bta[d];
  }
}

// ---------------- repack W[NH,D,HS] -> W'[D, NH*HS] ----------------
__global__ __launch_bounds__(256) void repack_qkv_kernel(const float* __restrict__ W,
                                                         float* __restrict__ out) {
  int tid = blockIdx.x * 256 + threadIdx.x;  // 0 .. D*D-1
  int d = tid >> 10;
  int n = tid & 1023;
  int h = n >> 6;
  int k = n & 63;
  out[tid] = W[(size_t)h * (Dd * HSs) + (size_t)d * HSs + k];
}

// ---------------- fp32 WMMA GEMM: C[M,N] = A[M,K] @ B[K,N] (+bias,+gelu,+residual) ----
// block tile 128(M) x 128(N), 256 threads = 8 waves, each wave 64x32 = 4x2 WMMA subtiles.
// Software-pipelined: global loads for tile k+1 overlap WMMA compute on tile k.
template <bool HAS_BIAS, bool HAS_RES, bool GELU>
__global__ __launch_bounds__(256) void gemm_kernel(const float* __restrict__ A,
                                                   const float* __restrict__ B,
                                                   float* __restrict__ C,
                                                   int M, int N, int K,
                                                   const float* __restrict__ bias,
                                                   const float* __restrict__ res) {
  __shared__ __align__(16) float As[128 * 20];   // padded stride 20 (16B aligned)
  __shared__ __align__(16) float Bs[16 * 132];   // padded stride 132

  int tid  = threadIdx.x;
  int lane = tid & 31;
  int wid  = tid >> 5;
  int lh   = lane & 15;
  int hi   = lane >> 4;
  int wm   = wid >> 2;   // 0..1 -> 64-row slab
  int wn   = wid & 3;    // 0..3 -> 32-col slab
  int m0   = blockIdx.y * 128;
  int n0   = blockIdx.x * 128;

  v8f acc[4][2];
#pragma unroll
  for (int i = 0; i < 4; ++i)
#pragma unroll
    for (int j = 0; j < 2; ++j)
      acc[i][j] = (v8f){0.f, 0.f, 0.f, 0.f, 0.f, 0.f, 0.f, 0.f};

  // preload k0 = 0
  float4 ra[2], rb[2];
#pragma unroll
  for (int i = 0; i < 2; ++i) {
    int idx4 = tid + i * 256;
    int row = idx4 >> 2, c4 = idx4 & 3;
    ra[i] = *(const float4*)(A + (size_t)(m0 + row) * K + c4 * 4);
  }
#pragma unroll
  for (int i = 0; i < 2; ++i) {
    int idx4 = tid + i * 256;
    int row = idx4 >> 5, c4 = idx4 & 31;
    rb[i] = *(const float4*)(B + (size_t)row * N + n0 + c4 * 4);
  }

  for (int k0 = 0; k0 < K; k0 += 16) {
    // stage current tile to LDS
#pragma unroll
    for (int i = 0; i < 2; ++i) {
      int idx4 = tid + i * 256;
      int row = idx4 >> 2, c4 = idx4 & 3;
      *(float4*)(&As[row * 20 + c4 * 4]) = ra[i];
    }
#pragma unroll
    for (int i = 0; i < 2; ++i) {
      int idx4 = tid + i * 256;
      int row = idx4 >> 5, c4 = idx4 & 31;
      *(float4*)(&Bs[row * 132 + c4 * 4]) = rb[i];
    }
    __syncthreads();

    // prefetch next tile (global only, overlaps WMMA below)
    int kn = k0 + 16;
    if (kn < K) {
#pragma unroll
      for (int i = 0; i < 2; ++i) {
        int idx4 = tid + i * 256;
        int row = idx4 >> 2, c4 = idx4 & 3;
        ra[i] = *(const float4*)(A + (size_t)(m0 + row) * K + kn + c4 * 4);
      }
#pragma unroll
      for (int i = 0; i < 2; ++i) {
        int idx4 = tid + i * 256;
        int row = idx4 >> 5, c4 = idx4 & 31;
        rb[i] = *(const float4*)(B + (size_t)(kn + row) * N + n0 + c4 * 4);
      }
    }

#pragma unroll
    for (int kk = 0; kk < 16; kk += 4) {
      v2f af[4], bf[2];
#pragma unroll
      for (int i = 0; i < 4; ++i) {
        int r = (wm * 64 + i * 16 + lh) * 20 + kk + hi * 2;
        af[i].x = As[r];
        af[i].y = As[r + 1];
      }
#pragma unroll
      for (int j = 0; j < 2; ++j) {
        int cidx = wn * 32 + j * 16 + lh;
        bf[j].x = Bs[(kk + hi * 2) * 132 + cidx];
        bf[j].y = Bs[(kk + hi * 2 + 1) * 132 + cidx];
      }
#pragma unroll
      for (int i = 0; i < 4; ++i)
#pragma unroll
        for (int j = 0; j < 2; ++j)
          acc[i][j] = WMMA_F32(af[i], bf[j], acc[i][j]);
    }
    __syncthreads();
  }

  // epilogue: C layout per ISA — VGPR r: lanes0-15 -> M=r, lanes16-31 -> M=8+r; N=lane%16
#pragma unroll
  for (int i = 0; i < 4; ++i) {
#pragma unroll
    for (int j = 0; j < 2; ++j) {
      int mb = m0 + wm * 64 + i * 16 + hi * 8;
      int nb = n0 + wn * 32 + j * 16 + lh;
      float bv = HAS_BIAS ? bias[nb] : 0.0f;
#pragma unroll
      for (int r = 0; r < 8; ++r) {
        float v = acc[i][j][r] + bv;
        if (GELU) v = gelu_exact(v);
        size_t off = (size_t)(mb + r) * N + nb;
        if (HAS_RES) v += res[off];
        C[off] = v;
      }
    }
  }
}

// ---------------- flash attention ----------------
// 4 waves / block, each wave owns one 16-row q block; K/V tiles shared in LDS.
// Every wave runs every kb tile; causal mask -inf makes out-of-range tiles an
// exact numerical no-op (p=0, corr=1), keeping barriers workgroup-uniform.
__global__ __launch_bounds__(128) void attn_kernel(const float* __restrict__ Q,
                                                   const float* __restrict__ Km,
                                                   const float* __restrict__ Vm,
                                                   float* __restrict__ O) {
  const int qg = blockIdx.x;   // group of 4 q-blocks
  const int h  = blockIdx.y;
  const int b  = blockIdx.z;

  __shared__ __align__(16) float Ks[16 * 68];
  __shared__ __align__(16) float Vs[16 * 68];
  __shared__ __align__(16) float Qs[4][16 * 68];
  __shared__ __align__(16) float Ps[4][16 * 17];

  int tid  = threadIdx.x;
  int lane = tid & 31;
  int wid  = tid >> 5;
  int lh   = lane & 15, hi = lane >> 4;
  int qb   = qg * 4 + wid;          // this wave's q block
  int kb_last = qg * 4 + 3;         // uniform loop bound for whole block

  const float* qbase = Q + ((size_t)(b * Tt + qb * 16)) * Dd + h * HSs;
#pragma unroll
  for (int j = 0; j < 8; ++j) {     // 16x64 = 256 float4 per wave
    int idx4 = lane + j * 32;
    int row = idx4 >> 4, c4 = idx4 & 15;
    *(float4*)(&Qs[wid][row * 68 + c4 * 4]) =
        *(const float4*)(qbase + (size_t)row * Dd + c4 * 4);
  }
  __syncthreads();

  v8f o0, o1, o2, o3;
  o0 = o1 = o2 = o3 = (v8f){0.f, 0.f, 0.f, 0.f, 0.f, 0.f, 0.f, 0.f};
  float m[8], l[8];
#pragma unroll
  for (int r = 0; r < 8; ++r) { m[r] = -INFINITY; l[r] = 0.f; }

  for (int kb = 0; kb <= kb_last; ++kb) {
    __syncthreads();
    const float* kbase = Km + ((size_t)(b * Tt + kb * 16)) * Dd + h * HSs;
    const float* vbase = Vm + ((size_t)(b * Tt + kb * 16)) * Dd + h * HSs;
#pragma unroll
    for (int j = 0; j < 2; ++j) {   // 256 float4 over 128 threads
      int idx4 = tid + j * 128;
      int row = idx4 >> 4, c4 = idx4 & 15;
      *(float4*)(&Ks[row * 68 + c4 * 4]) =
          *(const float4*)(kbase + (size_t)row * Dd + c4 * 4);
      *(float4*)(&Vs[row * 68 + c4 * 4]) =
          *(const float4*)(vbase + (size_t)row * Dd + c4 * 4);
    }
    __syncthreads();

    // S = q @ k^T  (16x16, K-dim = 64)
    v8f s = (v8f){0.f, 0.f, 0.f, 0.f, 0.f, 0.f, 0.f, 0.f};
#pragma unroll
    for (int kk = 0; kk < HSs; kk += 4) {
      v2f a, bb;
      int ka = kk + hi * 2;
      a.x  = Qs[wid][lh * 68 + ka]; a.y  = Qs[wid][lh * 68 + ka + 1];
      bb.x = Ks[lh * 68 + ka];      bb.y = Ks[lh * 68 + ka + 1];  // B[K][n] = k[n][K]
      s = WMMA_F32(a, bb, s);
    }

    float corr[8];
#pragma unroll
    for (int r = 0; r < 8; ++r) {
      int rowq = hi * 8 + r;
      float val = s[r] * 0.03125f;                   // scale = 1/sqrt(1024)
      if (kb * 16 + lh > qb * 16 + rowq) val = -INFINITY;  // causal mask (all cases)
      float mx = val;
#pragma unroll
      for (int off = 8; off >= 1; off >>= 1) mx = fmaxf(mx, __shfl_xor(mx, off, 32));
      float nm = fmaxf(m[r], mx);
      float p = expf(val - nm);
      float su = p;
#pragma unroll
      for (int off = 8; off >= 1; off >>= 1) su += __shfl_xor(su, off, 32);
      float c = expf(m[r] - nm);
      l[r] = l[r] * c + su;
      m[r] = nm;
      corr[r] = c;
      Ps[wid][rowq * 17 + lh] = p;
    }
#pragma unroll
    for (int r = 0; r < 8; ++r) {
      o0[r] *= corr[r]; o1[r] *= corr[r]; o2[r] *= corr[r]; o3[r] *= corr[r];
    }
    __syncthreads();

    // O += P @ V   (A = P 16x16, B = V 16x64)
#pragma unroll
    for (int kk = 0; kk < 16; kk += 4) {
      int ka = kk + hi * 2;
      v2f a;
      a.x = Ps[wid][lh * 17 + ka]; a.y = Ps[wid][lh * 17 + ka + 1];
      v2f b0, b1, b2, b3;
      b0.x = Vs[ka * 68 + 0  + lh]; b0.y = Vs[(ka + 1) * 68 + 0  + lh];
      b1.x = Vs[ka * 68 + 16 + lh]; b1.y = Vs[(ka + 1) * 68 + 16 + lh];
      b2.x = Vs[ka * 68 + 32 + lh]; b2.y = Vs[(ka + 1) * 68 + 32 + lh];
      b3.x = Vs[ka * 68 + 48 + lh]; b3.y = Vs[(ka + 1) * 68 + 48 + lh];
      o0 = WMMA_F32(a, b0, o0);
      o1 = WMMA_F32(a, b1, o1);
      o2 = WMMA_F32(a, b2, o2);
      o3 = WMMA_F32(a, b3, o3);
    }
  }

  float* obase = O + ((size_t)(b * Tt + qb * 16)) * Dd + h * HSs;
#pragma unroll
  for (int r = 0; r < 8; ++r) {
    int rowq = hi * 8 + r;
    float inv = 1.0f / l[r];
    obase[(size_t)rowq * Dd + 0  + lh] = o0[r] * inv;
    obase[(size_t)rowq * Dd + 16 + lh] = o1[r] * inv;
    obase[(size_t)rowq * Dd + 32 + lh] = o2[r] * inv;
    obase[(size_t)rowq * Dd + 48 + lh] = o3[r] * inv;
  }
}

// ---------------- per-row loss: -(logit[tgt] - logsumexp) ----------------
__global__ __launch_bounds__(256) void rowloss_kernel(const float* __restrict__ logits,
                                                      const int* __restrict__ tgt,
                                                      float* __restrict__ out) {
  int row = blockIdx.x;
  int tid = threadIdx.x;
  const float* lr = logits + (size_t)row * Vv;
  __shared__ float sh[256];
  float mx = -INFINITY;
  for (int i = tid; i < Vv; i += 256) mx = fmaxf(mx, lr[i]);
  sh[tid] = mx;
  __syncthreads();
  for (int off = 128; off > 0; off >>= 1) {
    if (tid < off) sh[tid] = fmaxf(sh[tid], sh[tid + off]);
    __syncthreads();
  }
  mx = sh[0];
  __syncthreads();
  float s = 0.f;
  for (int i = tid; i < Vv; i += 256) s += expf(lr[i] - mx);
  sh[tid] = s;
  __syncthreads();
  for (int off = 128; off > 0; off >>= 1) {
    if (tid < off) sh[tid] += sh[tid + off];
    __syncthreads();
  }
  if (tid == 0) out[row] = -(lr[tgt[row]] - mx - logf(sh[0]));
}

__global__ __launch_bounds__(256) void loss_reduce_kernel(const float* __restrict__ rowloss,
                                                          float* __restrict__ out) {
  __shared__ float sh[256];
  int tid = threadIdx.x;
  float s = 0.f;
  for (int i = tid; i < BT; i += 256) s += rowloss[i];
  sh[tid] = s;
  __syncthreads();
  for (int off = 128; off > 0; off >>= 1) {
    if (tid < off) sh[tid] += sh[tid + off];
    __syncthreads();
  }
  if (tid == 0) out[0] = sh[0] * (1.0f / BT);
}

// ---------------- host orchestration ----------------
static inline void launch_gemm(const float* A, const float* B, float* C, int M, int N, int K,
                               const float* bias, const float* res, int act,
                               hipStream_t stream) {
  dim3 grid(N / 128, M / 128);
  if (!bias && !res && act == 0)
    gemm_kernel<false, false, false><<<grid, 256, 0, stream>>>(A, B, C, M, N, K, bias, res);
  else if (bias && res && act == 0)
    gemm_kernel<true, true, false><<<grid, 256, 0, stream>>>(A, B, C, M, N, K, bias, res);
  else if (bias && !res && act == 1)
    gemm_kernel<true, false, true><<<grid, 256, 0, stream>>>(A, B, C, M, N, K, bias, res);
  else
    gemm_kernel<true, false, false><<<grid, 256, 0, stream>>>(A, B, C, M, N, K, bias, res);
}

extern "C" void kernel_launch(void* const* d_in, const int* in_sizes, int n_in,
                              void* d_out, int out_size, void* d_ws, size_t ws_size,
                              hipStream_t stream) {
  (void)in_sizes; (void)n_in; (void)out_size; (void)ws_size;
  const float* emb    = (const float*)d_in[0];
  const float* pose   = (const float*)d_in[1];
  const float* ln1_g  = (const float*)d_in[2];
  const float* ln1_b  = (const float*)d_in[3];
  const float* Wq     = (const float*)d_in[4];
  const float* Wk     = (const float*)d_in[5];
  const float* Wv     = (const float*)d_in[6];
  const float* Wo     = (const float*)d_in[7];
  const float* bo     = (const float*)d_in[8];
  const float* ln2_g  = (const float*)d_in[9];
  const float* ln2_b  = (const float*)d_in[10];
  const float* W1     = (const float*)d_in[11];
  const float* b1     = (const float*)d_in[12];
  const float* W2     = (const float*)d_in[13];
  const float* b2     = (const float*)d_in[14];
  const float* lnf_g  = (const float*)d_in[15];
  const float* lnf_b  = (const float*)d_in[16];
  const float* W_lm   = (const float*)d_in[17];
  const float* b_lm   = (const float*)d_in[18];
  const int*   x      = (const int*)d_in[19];
  const int*   tgt    = (const int*)d_in[20];

  const size_t MB = 1u << 20;
  char* w = (char*)d_ws;
  float* h   = (float*)(w + 0 * MB);     // 16 MB
  float* y   = (float*)(w + 16 * MB);    // 16 MB
  float* q   = (float*)(w + 32 * MB);    // 16 MB
  float* k   = (float*)(w + 48 * MB);    // 16 MB
  float* v   = (float*)(w + 64 * MB);    // 16 MB
  float* o   = (float*)(w + 80 * MB);    // 16 MB
  float* f1  = (float*)(w + 96 * MB);    // 64 MB
  float* wrp = (float*)(w + 160 * MB);   // 4 MB
  float* rls = (float*)(w + 164 * MB);   // 16 KB

  float* logits = (float*)d_out;
  float* lossp  = logits + (size_t)BT * Vv;

  embed_kernel<<<BT, 256, 0, stream>>>(emb, pose, x, h);

  for (int l = 0; l < Ll; ++l) {
    const size_t oW  = (size_t)l * Dd * Dd;
    const size_t oD  = (size_t)l * Dd;
    const size_t oF  = (size_t)l * Dd * FFD;
    const size_t oFb = (size_t)l * FFD;

    ln_kernel<<<BT, 256, 0, stream>>>(h, ln1_g + oD, ln1_b + oD, y);

    repack_qkv_kernel<<<(Dd * Dd) / 256, 256, 0, stream>>>(Wq + oW, wrp);
    launch_gemm(y, wrp, q, BT, Dd, Dd, nullptr, nullptr, 0, stream);
    repack_qkv_kernel<<<(Dd * Dd) / 256, 256, 0, stream>>>(Wk + oW, wrp);
    launch_gemm(y, wrp, k, BT, Dd, Dd, nullptr, nullptr, 0, stream);
    repack_qkv_kernel<<<(Dd * Dd) / 256, 256, 0, stream>>>(Wv + oW, wrp);
    launch_gemm(y, wrp, v, BT, Dd, Dd, nullptr, nullptr, 0, stream);

    attn_kernel<<<dim3(Tt / 64, NHh, Bb), 128, 0, stream>>>(q, k, v, o);

    // h = h + o @ Wo + bo
    launch_gemm(o, Wo + oW, h, BT, Dd, Dd, bo + oD, h, 0, stream);

    ln_kernel<<<BT, 256, 0, stream>>>(h, ln2_g + oD, ln2_b + oD, y);

    // f1 = gelu(y @ W1 + b1) ; h = h + f1 @ W2 + b2
    launch_gemm(y, W1 + oF, f1, BT, FFD, Dd, b1 + oFb, nullptr, 1, stream);
    launch_gemm(f1, W2 + oF, h, BT, Dd, FFD, b2 + oD, h, 0, stream);
  }

  ln_kernel<<<BT, 256, 0, stream>>>(h, lnf_g, lnf_b, y);
  launch_gemm(y, W_lm, logits, BT, Vv, Dd, b_lm, nullptr, 0, stream);

  rowloss_kernel<<<BT, 256, 0, stream>>>(logits, tgt, rls);
  loss_reduce_kernel<<<1, 256, 0, stream>>>(rls, lossp);
}